// GaussianMVNLL_87325275062293
// MI455X (gfx1250) — compile-verified
//
#include <hip/hip_runtime.h>
#include <math.h>

typedef __attribute__((ext_vector_type(2))) float v2f;
typedef __attribute__((ext_vector_type(8))) float v8f;

#define D_DIM 512
#define N_BATCH 256
#define EPS_DIAG 1e-6f

static __device__ __forceinline__ v8f wmma_f32_4(v2f a, v2f b, v8f c) {
  // V_WMMA_F32_16X16X4_F32 : D = A(16x4) * B(4x16) + C(16x16)
  return __builtin_amdgcn_wmma_f32_16x16x4_f32(false, a, false, b, (short)0, c,
                                               false, false);
}

// ---------------------------------------------------------------------------
// Kernel 1: batched SYRK  covs[b] = L[b] * L[b]^T + eps*I
// Symmetry-aware: grid (36, 256) over lower-triangular 64x64 tile-pairs.
// Block 128 (4 waves), 32x32 per wave. Global->LDS staging uses CDNA5 async
// loads (ASYNCcnt). Off-diagonal tiles are mirrored via an LDS transpose so
// both stores are coalesced.
// ---------------------------------------------------------------------------
__global__ __launch_bounds__(128) void syrk_kernel(const float* __restrict__ L,
                                                   float* __restrict__ covs) {
  // map pair index -> (bi >= bj)
  const int p = blockIdx.x;
  int bi = 0;
  while ((bi + 1) * (bi + 2) / 2 <= p) ++bi;
  const int bj = p - bi * (bi + 1) / 2;
  const int m0 = bi * 64;              // row block
  const int n0 = bj * 64;              // col block (n0 <= m0)
  const int b  = blockIdx.y;
  const bool diag = (m0 == n0);

  const float* Lb = L + (size_t)b * D_DIM * D_DIM;
  float* Cb = covs + (size_t)b * D_DIM * D_DIM;

  // 2 x (64 x 36) staging panels; reused as 64 x 68 transpose buffer
  __shared__ float smem[4608];
  float* As = smem;                    // 64 x 36 (stride 36: 16B rows, no conflicts)
  float* Bs = smem + 2304;             // 64 x 36
  const float* Bsel = diag ? As : Bs;  // diagonal pair: B-panel == A-panel

  const int tid  = threadIdx.x;
  const int wave = tid >> 5;
  const int lane = tid & 31;
  const int r = lane & 15;             // fragment row / C-fragment col
  const int h = lane >> 4;             // K-half (A/B), M-half (C)
  const int wr = wave >> 1, wc = wave & 1;
  const int ar0 = (32 * wr + r) * 36,      ar1 = (32 * wr + 16 + r) * 36;
  const int br0 = (32 * wc + r) * 36,      br1 = (32 * wc + 16 + r) * 36;

  v8f zero = {0.f, 0.f, 0.f, 0.f, 0.f, 0.f, 0.f, 0.f};
  v8f accs[2][2];
  accs[0][0] = zero; accs[0][1] = zero; accs[1][0] = zero; accs[1][1] = zero;

  for (int k0 = 0; k0 < D_DIM; k0 += 32) {
    // async global -> LDS staging (64 rows x 32 cols per panel, b128 per lane)
#pragma unroll
    for (int q = 0; q < 4; ++q) {
      int idx = tid + 128 * q;         // 0..511
      int row = idx >> 3;
      int c4  = (idx & 7) << 2;
      const float* ga = Lb + (size_t)(m0 + row) * D_DIM + k0 + c4;
      asm volatile("global_load_async_to_lds_b128 %0, %1, off"
                   :: "v"((unsigned)(size_t)&As[row * 36 + c4]), "v"(ga)
                   : "memory");
      if (!diag) {
        const float* gb = Lb + (size_t)(n0 + row) * D_DIM + k0 + c4;
        asm volatile("global_load_async_to_lds_b128 %0, %1, off"
                     :: "v"((unsigned)(size_t)&Bs[row * 36 + c4]), "v"(gb)
                     : "memory");
      }
    }
    asm volatile("s_wait_asynccnt 0x0" ::: "memory");
    __syncthreads();
#pragma unroll
    for (int kk = 0; kk < 32; kk += 4) {
      int kb = kk + 2 * h;             // ISA layout: K = 2*(lane/16) + vgpr
      v2f a0 = *(const v2f*)&As[ar0 + kb];
      v2f a1 = *(const v2f*)&As[ar1 + kb];
      v2f b0 = *(const v2f*)&Bsel[br0 + kb];
      v2f b1 = *(const v2f*)&Bsel[br1 + kb];
      accs[0][0] = wmma_f32_4(a0, b0, accs[0][0]);
      accs[0][1] = wmma_f32_4(a0, b1, accs[0][1]);
      accs[1][0] = wmma_f32_4(a1, b0, accs[1][0]);
      accs[1][1] = wmma_f32_4(a1, b1, accs[1][1]);
    }
    __syncthreads();
  }

  // direct store of the (m0,n0) tile (+eps on the diagonal)
  // C fragment: lanes 0-15 -> M=v, lanes 16-31 -> M=v+8; N = lane%16
#pragma unroll
  for (int a = 0; a < 2; ++a)
#pragma unroll
    for (int bq = 0; bq < 2; ++bq) {
      int R  = m0 + 32 * wr + 16 * a;
      int Cc = n0 + 32 * wc + 16 * bq;
      v8f acc = accs[a][bq];
#pragma unroll
      for (int v = 0; v < 8; ++v) {
        int row = R + v + 8 * h;
        int col = Cc + r;
        float val = acc[v];
        if (row == col) val += EPS_DIAG;
        Cb[(size_t)row * D_DIM + col] = val;
      }
    }

  if (!diag) {
    // mirror tile: transpose through LDS, then coalesced b128 stores
    float* Ts = smem;                  // 64 x 68 (stride 68: 16B rows)
#pragma unroll
    for (int a = 0; a < 2; ++a)
#pragma unroll
      for (int bq = 0; bq < 2; ++bq) {
        int Rl = 32 * wr + 16 * a;     // local row in tile
        int Cl = 32 * wc + 16 * bq;    // local col in tile
        v8f acc = accs[a][bq];
#pragma unroll
        for (int v = 0; v < 8; ++v)
          Ts[(Cl + r) * 68 + Rl + v + 8 * h] = acc[v];
      }
    __syncthreads();
#pragma unroll
    for (int q = 0; q < 8; ++q) {
      int idx = tid + 128 * q;         // 0..1023
      int row = idx >> 4;              // transposed row = original col
      int c4  = (idx & 15) << 2;
      float4 v4 = *(const float4*)&Ts[row * 68 + c4];
      *(float4*)(Cb + (size_t)(n0 + row) * D_DIM + m0 + c4) = v4;
    }
  }
}

// ---------------------------------------------------------------------------
// Kernel 2: blocked right-looking Cholesky, one WG (256 thr) per batch matrix.
// 16-wide panel factored in LDS; rank-16 trailing update via WMMA tiles.
// Only the lower triangle is kept valid.
// ---------------------------------------------------------------------------
__global__ __launch_bounds__(256) void chol_kernel(float* __restrict__ covs) {
  float* A = covs + (size_t)blockIdx.x * D_DIM * D_DIM;
  __shared__ float P[D_DIM * 18];     // panel: up to 512 rows x 16 cols, stride 18
  __shared__ float sdiag[2];          // [0]=d, [1]=1/d

  const int tid  = threadIdx.x;
  const int wave = tid >> 5;
  const int lane = tid & 31;
  const int r = lane & 15;
  const int h = lane >> 4;

  for (int j0 = 0; j0 < D_DIM; j0 += 16) {
    const int nr = D_DIM - j0;
    // load panel A[j0:512, j0:j0+16] into LDS
    for (int idx = tid; idx < nr * 16; idx += 256) {
      int rr = idx >> 4, cc = idx & 15;
      P[rr * 18 + cc] = A[(size_t)(j0 + rr) * D_DIM + j0 + cc];
    }
    __syncthreads();

    // factor the 16-column panel (reads lower triangle only)
    for (int jj = 0; jj < 16; ++jj) {
      if (tid == 0) {
        float d = sqrtf(P[jj * 18 + jj]);
        sdiag[0] = d;
        sdiag[1] = 1.0f / d;
      }
      __syncthreads();
      float dinv = sdiag[1];
      for (int rr = jj + tid; rr < nr; rr += 256)
        P[rr * 18 + jj] = (rr == jj) ? sdiag[0] : P[rr * 18 + jj] * dinv;
      __syncthreads();
      int w = 15 - jj;
      if (w > 0) {
        int tot = (nr - jj - 1) * w;
        for (int idx = tid; idx < tot; idx += 256) {
          int rr = jj + 1 + idx / w;
          int cc = jj + 1 + idx % w;
          P[rr * 18 + cc] -= P[rr * 18 + jj] * P[cc * 18 + jj];
        }
      }
      __syncthreads();
    }

    // write factored panel back
    for (int idx = tid; idx < nr * 16; idx += 256) {
      int rr = idx >> 4, cc = idx & 15;
      A[(size_t)(j0 + rr) * D_DIM + j0 + cc] = P[rr * 18 + cc];
    }

    // rank-16 trailing update on lower-triangular tiles:
    //   A[i,k] -= panel_i . panel_k^T   (16x16 tiles, 4 WMMA K-steps each)
    const int t0 = j0 + 16;
    const int nt = (D_DIM - t0) >> 4;
    if (nt > 0) {
      const int total = nt * (nt + 1) / 2;
      for (int t = wave; t < total; t += 8) {  // wave-uniform => EXEC all ones
        int ti = 0;
        while ((ti + 1) * (ti + 2) / 2 <= t) ++ti;
        int tk = t - ti * (ti + 1) / 2;
        int pi = 16 + ti * 16;                 // panel-relative row base
        int pk = 16 + tk * 16;
        v8f s = {0.f, 0.f, 0.f, 0.f, 0.f, 0.f, 0.f, 0.f};
#pragma unroll
        for (int kk = 0; kk < 16; kk += 4) {
          int kb = kk + 2 * h;
          v2f a  = *(const v2f*)&P[(pi + r) * 18 + kb];
          v2f bv = *(const v2f*)&P[(pk + r) * 18 + kb];
          s = wmma_f32_4(a, bv, s);
        }
        int gRow = j0 + pi, gCol = j0 + pk;
#pragma unroll
        for (int v = 0; v < 8; ++v) {
          size_t off = (size_t)(gRow + v + 8 * h) * D_DIM + gCol + r;
          A[off] -= s[v];
        }
      }
    }
    __syncthreads();
  }
}

// ---------------------------------------------------------------------------
// Kernel 3: per-batch forward substitution + logdet -> loss_b
//   loss_b = sum(log diag(Lc)) + 0.5*||Lc^{-1} diff||^2
// ---------------------------------------------------------------------------
__global__ __launch_bounds__(256) void solve_kernel(const float* __restrict__ covs,
                                                    const float* __restrict__ means,
                                                    const float* __restrict__ targets,
                                                    float* __restrict__ losses) {
  const int b = blockIdx.x;
  const float* A = covs + (size_t)b * D_DIM * D_DIM;
  __shared__ float y[D_DIM];
  __shared__ float red[256];
  const int tid = threadIdx.x;

  float ld = 0.0f;
  for (int i = tid; i < D_DIM; i += 256) {
    y[i] = targets[(size_t)b * D_DIM + i] - means[(size_t)b * D_DIM + i];
    ld += logf(A[(size_t)i * D_DIM + i]);   // 0.5*logdet = sum(log diag)
  }
  __syncthreads();

  // column-oriented forward substitution: Lc * y = diff
  for (int i = 0; i < D_DIM; ++i) {
    if (tid == 0) y[i] /= A[(size_t)i * D_DIM + i];
    __syncthreads();
    float yi = y[i];
    for (int j = i + 1 + tid; j < D_DIM; j += 256)
      y[j] -= A[(size_t)j * D_DIM + i] * yi;
    __syncthreads();
  }

  float q = 0.0f;
  for (int i = tid; i < D_DIM; i += 256) q += y[i] * y[i];
  red[tid] = ld + 0.5f * q;
  __syncthreads();
  for (int s = 128; s > 0; s >>= 1) {
    if (tid < s) red[tid] += red[tid + s];
    __syncthreads();
  }
  if (tid == 0) losses[b] = red[0];
}

// ---------------------------------------------------------------------------
// Kernel 4: mean over 256 batch losses -> d_out[0] (deterministic tree)
// ---------------------------------------------------------------------------
__global__ __launch_bounds__(256) void reduce_kernel(const float* __restrict__ losses,
                                                     float* __restrict__ out) {
  __shared__ float red[256];
  int tid = threadIdx.x;
  red[tid] = losses[tid];
  __syncthreads();
  for (int s = 128; s > 0; s >>= 1) {
    if (tid < s) red[tid] += red[tid + s];
    __syncthreads();
  }
  if (tid == 0) out[0] = red[0] * (1.0f / (float)N_BATCH);
}

// ---------------------------------------------------------------------------
extern "C" void kernel_launch(void* const* d_in, const int* in_sizes, int n_in,
                              void* d_out, int out_size, void* d_ws, size_t ws_size,
                              hipStream_t stream) {
  (void)in_sizes; (void)n_in; (void)out_size; (void)ws_size;
  const float* means   = (const float*)d_in[0];
  const float* targets = (const float*)d_in[1];
  const float* L       = (const float*)d_in[2];
  float* covs   = (float*)d_ws;                                   // 256*512*512 f32
  float* losses = covs + (size_t)N_BATCH * D_DIM * D_DIM;         // +256 f32
  float* out    = (float*)d_out;

  const int npairs = (D_DIM / 64) * (D_DIM / 64 + 1) / 2;         // 36
  dim3 g1(npairs, N_BATCH);
  syrk_kernel<<<g1, 128, 0, stream>>>(L, covs);
  chol_kernel<<<N_BATCH, 256, 0, stream>>>(covs);
  solve_kernel<<<N_BATCH, 256, 0, stream>>>(covs, means, targets, losses);
  reduce_kernel<<<1, 256, 0, stream>>>(losses, out);
}